// ImbGNN_20864951124664
// MI455X (gfx1250) — compile-verified
//
#include <hip/hip_runtime.h>
#include <math.h>

#define NNODES 100000
#define NEDGES 1600000
#define DIM    128
#define NCLS   10
#define NLAY   5
#define LSTR   132           // padded LDS row stride (floats) -> conflict-free
#define BN_EPS 1e-5f

typedef __attribute__((ext_vector_type(2))) float v2f;
typedef __attribute__((ext_vector_type(8))) float v8f;

__device__ __forceinline__ v8f wmma_f32(v2f a, v2f b, v8f c) {
  // V_WMMA_F32_16X16X4_F32 : D = A(16x4 f32) * B(4x16 f32) + C(16x16 f32)
  return __builtin_amdgcn_wmma_f32_16x16x4_f32(false, a, false, b,
                                               (short)0, c, false, false);
}

// ---------------------------------------------------------------- zero agg
__global__ __launch_bounds__(256) void zero_f4(float4* __restrict__ p) {
  p[(size_t)blockIdx.x * 256 + threadIdx.x] = make_float4(0.f, 0.f, 0.f, 0.f);
}

// ---------------------------------------------------------------- edge scatter
// tid -> (edge, 4-float chunk); 32 threads cover one edge's 128 features.
__global__ __launch_bounds__(256) void gin_scatter(const float* __restrict__ h,
                                                   const int*   __restrict__ ei,
                                                   float*       __restrict__ agg) {
  unsigned tid = blockIdx.x * 256u + threadIdx.x;
  unsigned e   = tid >> 5;
  unsigned c   = (tid & 31u) << 2;
  int s = ei[e];
  int d = ei[NEDGES + e];
  const float4 v = *reinterpret_cast<const float4*>(h + (unsigned)s * DIM + c);
  float* a = agg + (unsigned)d * DIM + c;
  unsafeAtomicAdd(a + 0, v.x);   // global_atomic_add_f32, no return
  unsafeAtomicAdd(a + 1, v.y);
  unsafeAtomicAdd(a + 2, v.z);
  unsafeAtomicAdd(a + 3, v.w);
}

// ---------------------------------------------------------------- fused GIN layer
// Block: 256 thr = 8 waves. 16 output rows per block, wave w owns columns
// [16w, 16w+16). GEMM1 -> BN -> ReLU -> LDS -> GEMM2 -> ReLU -> global.
__global__ __launch_bounds__(256) void gin_layer(const float* __restrict__ h,
                                                 const float* __restrict__ agg,
                                                 const float* __restrict__ W1,
                                                 const float* __restrict__ b1,
                                                 const float* __restrict__ gam,
                                                 const float* __restrict__ bet,
                                                 const float* __restrict__ mu,
                                                 const float* __restrict__ var,
                                                 const float* __restrict__ W2,
                                                 const float* __restrict__ b2,
                                                 float*       __restrict__ hout) {
  __shared__ float sh[16 * LSTR];
  __shared__ float sz[16 * LSTR];
  const int tid     = threadIdx.x;
  const int rowBase = blockIdx.x * 16;

  // stage (h + agg) tile: 16 x 128
  for (int i = tid; i < 16 * 32; i += 256) {
    int rr = i >> 5;
    int cc = (i & 31) << 2;
    int g  = (rowBase + rr) * DIM + cc;
    float4 a = *reinterpret_cast<const float4*>(h + g);
    float4 b = *reinterpret_cast<const float4*>(agg + g);
    float* dst = &sh[rr * LSTR + cc];
    dst[0] = a.x + b.x; dst[1] = a.y + b.y;
    dst[2] = a.z + b.z; dst[3] = a.w + b.w;
  }
  __syncthreads();

  const int wave = tid >> 5;
  const int lane = tid & 31;
  const int half = lane >> 4;
  const int r    = lane & 15;
  const int col  = wave * 16 + r;      // output column 0..127

  // ---- GEMM1: (h+agg) @ W1
  v8f acc = {};
  for (int kb = 0; kb < DIM; kb += 4) {
    int k0 = kb + 2 * half;
    v2f a, b;
    a.x = sh[r * LSTR + k0];
    a.y = sh[r * LSTR + k0 + 1];
    b.x = W1[k0 * DIM + col];
    b.y = W1[(k0 + 1) * DIM + col];
    acc = wmma_f32(a, b, acc);
  }
  // epilogue: +b1, BN(eval), ReLU -> LDS
  float sc = gam[col] * rsqrtf(var[col] + BN_EPS);
  float sf = bet[col] - mu[col] * sc;
  float bb = b1[col];
#pragma unroll
  for (int j = 0; j < 8; ++j) {
    float v = (acc[j] + bb) * sc + sf;
    sz[(j + 8 * half) * LSTR + col] = fmaxf(v, 0.f);
  }
  __syncthreads();

  // ---- GEMM2: z @ W2
  v8f acc2 = {};
  for (int kb = 0; kb < DIM; kb += 4) {
    int k0 = kb + 2 * half;
    v2f a, b;
    a.x = sz[r * LSTR + k0];
    a.y = sz[r * LSTR + k0 + 1];
    b.x = W2[k0 * DIM + col];
    b.y = W2[(k0 + 1) * DIM + col];
    acc2 = wmma_f32(a, b, acc2);
  }
  float c2 = b2[col];
#pragma unroll
  for (int j = 0; j < 8; ++j) {
    hout[(rowBase + j + 8 * half) * DIM + col] = fmaxf(acc2[j] + c2, 0.f);
  }
}

// ---------------------------------------------------------------- head: lin1+relu, lin2, log_softmax
__global__ __launch_bounds__(256) void gin_head(const float* __restrict__ h,
                                                const float* __restrict__ w1,
                                                const float* __restrict__ b1v,
                                                const float* __restrict__ w2,
                                                const float* __restrict__ b2v,
                                                float*       __restrict__ out) {
  __shared__ float sh[16 * LSTR];
  __shared__ float sz[16 * LSTR];
  __shared__ float ss[16 * 16];
  const int tid     = threadIdx.x;
  const int rowBase = blockIdx.x * 16;

  for (int i = tid; i < 16 * 32; i += 256) {
    int rr = i >> 5;
    int cc = (i & 31) << 2;
    float4 a = *reinterpret_cast<const float4*>(h + (rowBase + rr) * DIM + cc);
    float* dst = &sh[rr * LSTR + cc];
    dst[0] = a.x; dst[1] = a.y; dst[2] = a.z; dst[3] = a.w;
  }
  __syncthreads();

  const int wave = tid >> 5;
  const int lane = tid & 31;
  const int half = lane >> 4;
  const int r    = lane & 15;
  const int col  = wave * 16 + r;

  // lin1 (128x128) + relu
  v8f acc = {};
  for (int kb = 0; kb < DIM; kb += 4) {
    int k0 = kb + 2 * half;
    v2f a, b;
    a.x = sh[r * LSTR + k0];
    a.y = sh[r * LSTR + k0 + 1];
    b.x = w1[k0 * DIM + col];
    b.y = w1[(k0 + 1) * DIM + col];
    acc = wmma_f32(a, b, acc);
  }
  float bb = b1v[col];
#pragma unroll
  for (int j = 0; j < 8; ++j)
    sz[(j + 8 * half) * LSTR + col] = fmaxf(acc[j] + bb, 0.f);
  __syncthreads();

  // lin2 (128x10), N padded to 16; wave 0 only
  if (wave == 0) {
    v8f a2 = {};
    for (int kb = 0; kb < DIM; kb += 4) {
      int k0 = kb + 2 * half;
      v2f a, b;
      a.x = sz[r * LSTR + k0];
      a.y = sz[r * LSTR + k0 + 1];
      b.x = (r < NCLS) ? w2[k0 * NCLS + r] : 0.f;
      b.y = (r < NCLS) ? w2[(k0 + 1) * NCLS + r] : 0.f;
      a2 = wmma_f32(a, b, a2);
    }
    float c2 = (r < NCLS) ? b2v[r] : 0.f;
#pragma unroll
    for (int j = 0; j < 8; ++j)
      ss[(j + 8 * half) * 16 + r] = a2[j] + c2;
  }
  __syncthreads();

  // log_softmax over 10 classes, one thread per row
  if (tid < 16) {
    const float* row = &ss[tid * 16];
    float m = row[0];
    for (int n = 1; n < NCLS; ++n) m = fmaxf(m, row[n]);
    float s = 0.f;
    for (int n = 0; n < NCLS; ++n) s += __expf(row[n] - m);
    float lse = m + __logf(s);
    float* o = out + (size_t)(rowBase + tid) * NCLS;
    for (int n = 0; n < NCLS; ++n) o[n] = row[n] - lse;
  }
}

// ---------------------------------------------------------------- launch
extern "C" void kernel_launch(void* const* d_in, const int* in_sizes, int n_in,
                              void* d_out, int out_size, void* d_ws, size_t ws_size,
                              hipStream_t stream) {
  (void)in_sizes; (void)n_in; (void)out_size; (void)ws_size;
  const float* x    = (const float*)d_in[0];
  const int*   ei   = (const int*)  d_in[1];
  const float* W1   = (const float*)d_in[2];
  const float* b1   = (const float*)d_in[3];
  const float* gam  = (const float*)d_in[4];
  const float* bet  = (const float*)d_in[5];
  const float* mu   = (const float*)d_in[6];
  const float* var  = (const float*)d_in[7];
  const float* W2   = (const float*)d_in[8];
  const float* b2   = (const float*)d_in[9];
  const float* l1w  = (const float*)d_in[10];
  const float* l1b  = (const float*)d_in[11];
  const float* l2w  = (const float*)d_in[12];
  const float* l2b  = (const float*)d_in[13];
  float* out = (float*)d_out;

  const size_t NH = (size_t)NNODES * DIM;          // 12.8M floats
  float* hA  = (float*)d_ws;
  float* hB  = hA + NH;
  float* agg = hA + 2 * NH;

  const float* hcur = x;
  float* bufs[2] = {hA, hB};
  for (int l = 0; l < NLAY; ++l) {
    zero_f4<<<(NH / 4) / 256, 256, 0, stream>>>((float4*)agg);           // 12500 blocks
    gin_scatter<<<(NEDGES * 32) / 256, 256, 0, stream>>>(hcur, ei, agg); // 200000 blocks
    float* hnext = bufs[l & 1];
    gin_layer<<<NNODES / 16, 256, 0, stream>>>(hcur, agg,
        W1 + (size_t)l * DIM * DIM, b1 + l * DIM,
        gam + l * DIM, bet + l * DIM, mu + l * DIM, var + l * DIM,
        W2 + (size_t)l * DIM * DIM, b2 + l * DIM, hnext);
    hcur = hnext;
  }
  gin_head<<<NNODES / 16, 256, 0, stream>>>(hcur, l1w, l1b, l2w, l2b, out);
}